// Net_48249662603297
// MI455X (gfx1250) — compile-verified
//
#include <hip/hip_runtime.h>

// GCN link-predictor for MI455X (gfx1250, wave32).
// GEMMs use V_WMMA_F32_16X16X4_F32 (exact fp32, matches f32 reference).
// Edge aggregation uses native no-return global_atomic_add_f32 (inline asm)
// into L2-resident accumulators (z1/z2 fit easily in the 192MB L2).

#define NNODES 100000

typedef __attribute__((ext_vector_type(2))) float v2f;
typedef __attribute__((ext_vector_type(8))) float v8f;

// Native fire-and-forget f32 atomic add (no return -> STOREcnt, no CAS loop).
// s_endpgm performs an implicit wait-idle, so no explicit storecnt wait needed.
__device__ __forceinline__ void atomic_add_f32_nr(float* p, float v) {
  asm volatile("global_atomic_add_f32 %0, %1, off" :: "v"(p), "v"(v) : "memory");
}

// ---------------- degree / normalization ----------------

__global__ void k_deg_init(unsigned* __restrict__ deg, int n) {
  int i = blockIdx.x * blockDim.x + threadIdx.x;
  if (i < n) deg[i] = 1u;                   // self-loop
}

__global__ void k_deg_accum(const int* __restrict__ dst, unsigned* __restrict__ deg, int E) {
  int e = blockIdx.x * blockDim.x + threadIdx.x;
  if (e < E) atomicAdd(&deg[dst[e]], 1u);   // integer atomic: always native
}

__global__ void k_dinv(unsigned* __restrict__ degU, float* __restrict__ dinv, int n) {
  int i = blockIdx.x * blockDim.x + threadIdx.x;
  if (i < n) {
    float d = (float)degU[i];               // exact (deg << 2^24)
    dinv[i] = 1.0f / sqrtf(d);              // deg >= 1 always
  }
}

// ---------------- WMMA GEMM: H[N,NOUT] = X[N,128] * W[128,NOUT] ----------------
// grid.x = N/16 row tiles (100000/16 = 6250 exact), block = (NOUT/16) waves.
// Fragment layouts per CDNA5 ISA 7.12.2 (f32):
//   A 16x4 : lane&15 = M row; lane>>4 selects K pair {0,1} or {2,3}; 2 VGPRs = K,K+1
//   B 4x16 : lane&15 = N col; VGPR v holds row K = v + 2*(lane>>4)
//   C/D    : lane&15 = N col; VGPR r holds row M = r + 8*(lane>>4)

template<int NOUT>
__global__ void k_gemm_wmma(const float* __restrict__ X, const float* __restrict__ W,
                            float* __restrict__ H) {
  constexpr int NW = NOUT / 16;            // waves per block
  __shared__ float As[16 * 130];           // pad 128->130: conflict-free stride-130 reads
  const int tid = threadIdx.x;
  const int m0  = blockIdx.x * 16;

  // cooperative load of the 16x128 A tile (float4 global reads, padded LDS rows)
  for (int i = tid * 4; i < 16 * 128; i += NW * 32 * 4) {
    float4 v = *(const float4*)(X + (size_t)m0 * 128 + i);
    int r = i >> 7, c = i & 127;
    float* p = &As[r * 130 + c];
    p[0] = v.x; p[1] = v.y; p[2] = v.z; p[3] = v.w;
  }
  __syncthreads();

  const int wave = tid >> 5;
  const int lane = tid & 31;
  const int hi   = lane >> 4;              // 0|1 : which K/M half this lane holds
  const int lo   = lane & 15;
  const int n0   = wave * 16;

  v8f c = {};
#pragma unroll
  for (int k0 = 0; k0 < 128; k0 += 4) {
    const int kk = k0 + 2 * hi;
    v2f a = *(const v2f*)&As[lo * 130 + kk];          // A[m=lo][kk], A[m=lo][kk+1]
    v2f b;
    b.x = W[(size_t)kk       * NOUT + n0 + lo];       // B[kk  ][n=lo]
    b.y = W[(size_t)(kk + 1) * NOUT + n0 + lo];       // B[kk+1][n=lo]
    c = __builtin_amdgcn_wmma_f32_16x16x4_f32(
        /*neg_a=*/false, a, /*neg_b=*/false, b,
        /*c_mod=*/(short)0, c, /*reuse_a=*/false, /*reuse_b=*/false);
  }

  float* Hp = H + (size_t)m0 * NOUT + n0 + lo;
#pragma unroll
  for (int r = 0; r < 8; ++r)
    Hp[(size_t)(r + 8 * hi) * NOUT] = c[r];
}

// ---------------- edge aggregation: z[dst] += h[src] * dinv[src]*dinv[dst] ----------------
// One wave per edge: lane loads F/32 contiguous floats of the source row
// (b128 for F=128, b64 for F=64), then fires F/32 no-return f32 atomics.

template<int F>
__global__ void k_edge_agg(const int* __restrict__ src, const int* __restrict__ dst,
                           const float* __restrict__ h, const float* __restrict__ dinv,
                           float* __restrict__ z, int E) {
  constexpr int VEC = F / 32;               // floats per lane (4 or 2)
  const int wpb = blockDim.x >> 5;
  int e = blockIdx.x * wpb + (threadIdx.x >> 5);
  if (e >= E) return;
  int lane = threadIdx.x & 31;
  int s = src[e], d = dst[e];               // wave-uniform -> broadcast loads
  float norm = dinv[s] * dinv[d];

  const float* hrow = h + (size_t)s * F + lane * VEC;
  float*       zrow = z + (size_t)d * F + lane * VEC;

  float vals[VEC];
  if constexpr (VEC == 4) {
    float4 v = *(const float4*)hrow;        // global_load_b128
    vals[0] = v.x; vals[1] = v.y; vals[2] = v.z; vals[3] = v.w;
  } else {
    float2 v = *(const float2*)hrow;        // global_load_b64
    vals[0] = v.x; vals[1] = v.y;
  }
#pragma unroll
  for (int j = 0; j < VEC; ++j)
    atomic_add_f32_nr(zrow + j, vals[j] * norm);
}

// ---------------- finalize: z = [relu](z + h*dinv^2 + bias) ----------------

template<int F, bool RELU>
__global__ void k_finalize(const float* __restrict__ h, const float* __restrict__ dinv,
                           const float* __restrict__ bias, float* __restrict__ z,
                           long total) {
  long idx = (long)blockIdx.x * blockDim.x + threadIdx.x;
  if (idx >= total) return;
  int i = (int)(idx / F);
  int f = (int)(idx & (F - 1));
  float di = dinv[i];
  float v = z[idx] + h[idx] * di * di + bias[f];
  z[idx] = RELU ? fmaxf(v, 0.0f) : v;
}

// ---------------- decode: logits[p] = dot(z2[a], z2[b]) over 64 dims ----------------

__global__ void k_decode(const int* __restrict__ pa, const int* __restrict__ pb,
                         const int* __restrict__ na, const int* __restrict__ nb,
                         const float* __restrict__ z, float* __restrict__ out,
                         int EP, int EN) {
  int pair = blockIdx.x * (blockDim.x >> 5) + (threadIdx.x >> 5);
  if (pair >= EP + EN) return;
  int a, b;
  if (pair < EP) { a = pa[pair]; b = pb[pair]; }
  else           { a = na[pair - EP]; b = nb[pair - EP]; }
  int lane = threadIdx.x & 31;
  const float* za = z + (size_t)a * 64;
  const float* zb = z + (size_t)b * 64;
  float acc = za[lane] * zb[lane] + za[lane + 32] * zb[lane + 32];
#pragma unroll
  for (int off = 16; off; off >>= 1)
    acc += __shfl_xor(acc, off, 32);        // wave32 reduction
  if (lane == 0) out[pair] = acc;
}

// ---------------- launcher ----------------

extern "C" void kernel_launch(void* const* d_in, const int* in_sizes, int n_in,
                              void* d_out, int out_size, void* d_ws, size_t ws_size,
                              hipStream_t stream) {
  const float* x  = (const float*)d_in[0];
  const int*   ei = (const int*)  d_in[1];
  const int*   pos= (const int*)  d_in[2];
  const int*   neg= (const int*)  d_in[3];
  const float* W1 = (const float*)d_in[4];
  const float* b1 = (const float*)d_in[5];
  const float* W2 = (const float*)d_in[6];
  const float* b2 = (const float*)d_in[7];

  const int N  = NNODES;
  const int E  = in_sizes[1] / 2;
  const int EP = in_sizes[2] / 2;
  const int EN = in_sizes[3] / 2;
  const int* src = ei;
  const int* dst = ei + E;

  // workspace layout (~128.8 MB total)
  unsigned* degU = (unsigned*)d_ws;         // N (u32 degree counts)
  float* dinv = (float*)d_ws + N;           // N
  float* h1   = dinv + N;                   // N*128 ; reused as h2 (N*64) after layer 1
  float* z1   = h1 + (size_t)N * 128;       // N*128
  float* z2   = z1 + (size_t)N * 128;       // N*64
  float* h2   = h1;

  hipMemsetAsync(z1, 0, (size_t)N * 128 * sizeof(float), stream);
  hipMemsetAsync(z2, 0, (size_t)N * 64  * sizeof(float), stream);

  // symmetric normalization coefficients (exact integer degree counting)
  k_deg_init <<<(N + 255) / 256, 256, 0, stream>>>(degU, N);
  k_deg_accum<<<(E + 255) / 256, 256, 0, stream>>>(dst, degU, E);
  k_dinv     <<<(N + 255) / 256, 256, 0, stream>>>(degU, dinv, N);

  // layer 1: h1 = x@W1 ; z1 = relu(scatter(h1) + selfloop + b1)
  k_gemm_wmma<128><<<N / 16, 256, 0, stream>>>(x, W1, h1);
  k_edge_agg<128> <<<(E + 7) / 8, 256, 0, stream>>>(src, dst, h1, dinv, z1, E);
  k_finalize<128, true><<<(int)(((long)N * 128 + 255) / 256), 256, 0, stream>>>(
      h1, dinv, b1, z1, (long)N * 128);

  // layer 2: h2 = z1@W2 ; z2 = scatter(h2) + selfloop + b2
  k_gemm_wmma<64><<<N / 16, 128, 0, stream>>>(z1, W2, h2);
  k_edge_agg<64> <<<(E + 7) / 8, 256, 0, stream>>>(src, dst, h2, dinv, z2, E);
  k_finalize<64, false><<<(int)(((long)N * 64 + 255) / 256), 256, 0, stream>>>(
      h2, dinv, b2, z2, (long)N * 64);

  // decode: dot products over pos ++ neg pairs
  k_decode<<<(EP + EN + 7) / 8, 256, 0, stream>>>(
      pos, pos + EP, neg, neg + EN, z2, (float*)d_out, EP, EN);
}